// DecoderSelfAttention_37855841747717
// MI455X (gfx1250) — compile-verified
//
#include <hip/hip_runtime.h>
#include <math.h>

// Problem constants (from reference): B=16, Q=1, S=2048, E=4096, H=32, C=128
#define Bb 16
#define Ss 2048
#define Ee 4096
#define Hh 32
#define Cc 128

typedef __attribute__((ext_vector_type(2))) float v2f;
typedef __attribute__((ext_vector_type(8))) float v8f;
typedef int v4i __attribute__((vector_size(16)));   // matches builtin param type

// CDNA5 async global->LDS path (ASYNCcnt-tracked), if this toolchain declares it.
#if defined(__has_builtin)
#  if __has_builtin(__builtin_amdgcn_global_load_async_to_lds_b128) && \
      __has_builtin(__builtin_amdgcn_s_wait_asynccnt)
#    define USE_ASYNC_LDS 1
#  endif
#endif
#ifndef USE_ASYNC_LDS
#  define USE_ASYNC_LDS 0
#endif

// ---------------------------------------------------------------------------
// OUT[16, 4096] = X[16, 4096] @ W[4096, 4096]^T   (out[m,n] = sum_k X[m,k]*W[n,k])
// One wave per 16x16 output tile, V_WMMA_F32_16X16X4_F32 over K in chunks of 64
// staged through LDS (W chunk via async-to-LDS when available). 8 waves cover
// 128 N-columns per block; grid.x = 4096/128 = 32.
// ---------------------------------------------------------------------------
__global__ __launch_bounds__(256) void gemm16_nt_wmma(const float* __restrict__ X,
                                                      const float* __restrict__ W,
                                                      float* __restrict__ OUT) {
  const int K = Ee, N = Ee;
  const int KC = 64;
  __shared__ __align__(16) float xs[16][68];        // X chunk  [m][k]
  __shared__ __align__(16) float wsm[8][16][68];    // per-wave W chunk [n][k]

  const int tid  = threadIdx.x;
  const int lane = tid & 31;
  const int wave = tid >> 5;
  const int half = lane >> 4;   // 0: K pair base 0 ; 1: K pair base 2
  const int lid  = lane & 15;   // M row (A) / N col (B,C,D)
  const int n0   = blockIdx.x * 128 + wave * 16;

  v8f acc = {};

  for (int k0 = 0; k0 < K; k0 += KC) {
    // --- stage X chunk [16, 64]: 256 threads x one float4 each, coalesced ---
    {
      const int m  = tid >> 4;
      const int kk = (tid & 15) << 2;
      const float4 xv = *(const float4*)(X + (size_t)m * K + k0 + kk);
      xs[m][kk + 0] = xv.x; xs[m][kk + 1] = xv.y;
      xs[m][kk + 2] = xv.z; xs[m][kk + 3] = xv.w;
    }
    // --- stage this wave's W chunk [16 rows, 64 k]: 8 x 16B per lane ---
    #pragma unroll
    for (int r = 0; r < 8; ++r) {
      const int f   = r * 32 + lane;
      const int row = f >> 4;
      const int kq  = (f & 15) << 2;
      const float* gsrc = W + (size_t)(n0 + row) * K + k0 + kq;
      float*       ldst = &wsm[wave][row][kq];
#if USE_ASYNC_LDS
      __builtin_amdgcn_global_load_async_to_lds_b128(
          (__attribute__((address_space(1))) v4i*)gsrc,
          (__attribute__((address_space(3))) v4i*)ldst,
          0, 0);
#else
      const float4 wv = *(const float4*)gsrc;
      ldst[0] = wv.x; ldst[1] = wv.y; ldst[2] = wv.z; ldst[3] = wv.w;
#endif
    }
    // prefetch next weight chunk into the cache hierarchy
    if (k0 + KC < K) {
      __builtin_prefetch(W + (size_t)(n0 + lid) * K + k0 + KC + half * 32, 0, 0);
    }
#if USE_ASYNC_LDS
    __builtin_amdgcn_s_wait_asynccnt(0);
#endif
    __syncthreads();

    // --- 16 WMMAs per chunk ---
    #pragma unroll
    for (int kk = 0; kk < KC; kk += 4) {
      // A 16x4 f32: lanes 0-15 -> M=lid, K={kk,kk+1}; lanes 16-31 -> K={kk+2,kk+3}
      v2f a, bfrag;
      a.x = xs[lid][kk + 2 * half];
      a.y = xs[lid][kk + 2 * half + 1];
      // B 4x16 f32: B[k][n] = W[n][k], same striping with N=lid
      bfrag.x = wsm[wave][lid][kk + 2 * half];
      bfrag.y = wsm[wave][lid][kk + 2 * half + 1];
      acc = __builtin_amdgcn_wmma_f32_16x16x4_f32(
          /*neg_a=*/false, a, /*neg_b=*/false, bfrag,
          /*c_mod=*/(short)0, acc, /*reuse_a=*/false, /*reuse_b=*/false);
    }
    __syncthreads();
  }

  // C/D layout: VGPR j -> M=j (lanes 0-15), M=j+8 (lanes 16-31); N = lid
  #pragma unroll
  for (int j = 0; j < 8; ++j) {
    const int row = half * 8 + j;
    OUT[(size_t)row * N + n0 + lid] = acc[j];
  }
}

// ---------------------------------------------------------------------------
// Overwrite cache row `token_index` in the copied output cache with new K/V.
// grid.x = 2B = 32 (kb), 256 threads stream the 4096-float row.
// ---------------------------------------------------------------------------
__global__ __launch_bounds__(256) void scatter_kv(const float* __restrict__ kproj,
                                                  const float* __restrict__ vproj,
                                                  const int* __restrict__ tok,
                                                  float* __restrict__ out_attn) {
  const int kb = blockIdx.x;            // 0..31
  const int t  = tok[0];
  const float* src = (kb < Bb) ? (kproj + (size_t)kb * Ee)
                               : (vproj + (size_t)(kb - Bb) * Ee);
  float* dst = out_attn + ((size_t)kb * Ss + (size_t)t) * (size_t)Ee;
  for (int e = threadIdx.x; e < Ee; e += 256) dst[e] = src[e];
}

// ---------------------------------------------------------------------------
// One workgroup per (b, h): scores -> LDS, stable softmax, weighted-V sum.
// Reads the ORIGINAL cache, substituting fresh K/V at s == token_index.
// ---------------------------------------------------------------------------
__global__ __launch_bounds__(256) void attn_kernel(const float* __restrict__ qproj,
                                                   const float* __restrict__ kproj,
                                                   const float* __restrict__ vproj,
                                                   const float* __restrict__ cache,
                                                   const unsigned char* __restrict__ mask,
                                                   const int* __restrict__ tok,
                                                   float* __restrict__ ao) {
  const int b   = blockIdx.x >> 5;      // / H
  const int h   = blockIdx.x & 31;      // % H
  const int tid = threadIdx.x;

  __shared__ float sc[Ss];                       // 8 KB of scores/probs
  __shared__ __align__(16) float qs[Cc];
  __shared__ float red[256];

  const int t = tok[0];
  if (tid < Cc) qs[tid] = qproj[(size_t)b * Ee + h * Cc + tid];
  __syncthreads();

  const float scale = 0.08838834764831845f;      // 1/sqrt(128)
  const float4* q4 = (const float4*)qs;

  // --- scores ---
  for (int s = tid; s < Ss; s += 256) {
    const float* kp = (s == t) ? (kproj + (size_t)b * Ee + h * Cc)
                               : (cache + ((size_t)b * Ss + s) * Ee + h * Cc);
    const float4* k4 = (const float4*)kp;
    float d = 0.f;
    #pragma unroll 8
    for (int i = 0; i < Cc / 4; ++i) {
      const float4 kv = k4[i];
      const float4 qv = q4[i];
      d += kv.x * qv.x + kv.y * qv.y + kv.z * qv.z + kv.w * qv.w;
    }
    sc[s] = d * scale + (mask[(size_t)b * Ss + s] ? 0.f : -1e12f);
  }
  __syncthreads();

  // --- block max ---
  float lm = -INFINITY;
  for (int s = tid; s < Ss; s += 256) lm = fmaxf(lm, sc[s]);
  red[tid] = lm; __syncthreads();
  for (int off = 128; off > 0; off >>= 1) {
    if (tid < off) red[tid] = fmaxf(red[tid], red[tid + off]);
    __syncthreads();
  }
  const float mx = red[0]; __syncthreads();

  // --- exp + block sum ---
  float ls = 0.f;
  for (int s = tid; s < Ss; s += 256) {
    const float p = __expf(sc[s] - mx);
    sc[s] = p;
    ls += p;
  }
  red[tid] = ls; __syncthreads();
  for (int off = 128; off > 0; off >>= 1) {
    if (tid < off) red[tid] += red[tid + off];
    __syncthreads();
  }
  const float inv = 1.0f / red[0]; __syncthreads();

  // --- weighted V: 2 s-groups x 128 channels, coalesced per s-row ---
  const int c   = tid & (Cc - 1);
  const int grp = tid >> 7;
  float acc = 0.f;
  for (int s = grp; s < Ss; s += 2) {
    const float* vp = (s == t) ? (vproj + (size_t)b * Ee + h * Cc)
                               : (cache + ((size_t)(Bb + b) * Ss + s) * Ee + h * Cc);
    acc += sc[s] * vp[c];
  }
  red[tid] = acc; __syncthreads();
  if (grp == 0)
    ao[(size_t)b * Ee + h * Cc + c] = (red[c] + red[128 + c]) * inv;
}

// ---------------------------------------------------------------------------
extern "C" void kernel_launch(void* const* d_in, const int* in_sizes, int n_in,
                              void* d_out, int out_size, void* d_ws, size_t ws_size,
                              hipStream_t stream) {
  const float*         dec   = (const float*)d_in[0];          // [16, 1, 4096]
  const float*         cache = (const float*)d_in[1];          // [32, 2048, 4096]
  const unsigned char* mask  = (const unsigned char*)d_in[2];  // [16, 2048] bool
  const int*           tok   = (const int*)d_in[3];            // [1]
  const float*         Wq    = (const float*)d_in[4];
  const float*         Wk    = (const float*)d_in[5];
  const float*         Wv    = (const float*)d_in[6];
  const float*         Wo    = (const float*)d_in[7];

  float* out_dec  = (float*)d_out;                                   // 16*4096
  float* out_attn = (float*)d_out + (size_t)Bb * Ee;                 // 32*2048*4096

  float* ws = (float*)d_ws;
  float* q  = ws;                       // 65536 floats
  float* k  = ws + (size_t)Bb * Ee;     // 65536
  float* v  = ws + (size_t)2 * Bb * Ee; // 65536
  float* ao = ws + (size_t)3 * Bb * Ee; // 65536

  // 1 GiB cache copy to output via DMA engine (graph-capturable)
  (void)hipMemcpyAsync(out_attn, cache,
                       (size_t)2 * Bb * Ss * Ee * sizeof(float),
                       hipMemcpyDeviceToDevice, stream);

  // Q/K/V projections (WMMA f32)
  gemm16_nt_wmma<<<dim3(Ee / 128), dim3(256), 0, stream>>>(dec, Wq, q);
  gemm16_nt_wmma<<<dim3(Ee / 128), dim3(256), 0, stream>>>(dec, Wk, k);
  gemm16_nt_wmma<<<dim3(Ee / 128), dim3(256), 0, stream>>>(dec, Wv, v);

  // scatter fresh K/V into copied cache at token_index
  scatter_kv<<<dim3(2 * Bb), dim3(256), 0, stream>>>(k, v, tok, out_attn);

  // attention (reads original cache + fresh K/V at token_index)
  attn_kernel<<<dim3(Bb * Hh), dim3(256), 0, stream>>>(q, k, v, cache, mask, tok, ao);

  // output projection
  gemm16_nt_wmma<<<dim3(Ee / 128), dim3(256), 0, stream>>>(ao, Wo, out_dec);
}